// MandelbulbNode_936302871030
// MI455X (gfx1250) — compile-verified
//
#include <hip/hip_runtime.h>
#include <hip/hip_bf16.h>

typedef __attribute__((ext_vector_type(2))) float v2f;
typedef __attribute__((ext_vector_type(8))) float v8f;

#define EPSV 1e-8f

// ---------------------------------------------------------------------------
// Kernel 1: Mandelbulb distance estimator, transcendental-free inner loop.
// POWER=8 -> r^8 by squaring; sin/cos(8*angle) by three complex squarings of
// (cos,sin) obtained algebraically. Only v_sqrt/v_rcp/v_rsq per iteration.
// ---------------------------------------------------------------------------
__global__ void __launch_bounds__(256)
mandelbulb_dist_kernel(const float* __restrict__ x, float* __restrict__ dist, int n) {
    int i = blockIdx.x * blockDim.x + threadIdx.x;
    if (i >= n) return;

    const float* xi = x + 3ll * i;       // contiguous per-lane 12B -> b96 coalesced
    float px = xi[0], py = xi[1], pz = xi[2];

    float zx = px, zy = py, zz = pz;
    float dr = 1.0f;

#pragma unroll
    for (int it = 0; it < 8; ++it) {
        float r2 = __builtin_fmaf(zx, zx, __builtin_fmaf(zy, zy, zz * zz));
        float r  = __builtin_amdgcn_sqrtf(r2);
        bool esc = r > 2.0f;

        // cos(theta) = clamp(z/(r+eps)), sin(theta) = sqrt(1-c^2) >= 0
        float c = zz * __builtin_amdgcn_rcpf(r + EPSV);
        c = fminf(1.0f, fmaxf(-1.0f, c));
        float s = __builtin_amdgcn_sqrtf(fmaxf(0.0f, __builtin_fmaf(-c, c, 1.0f)));
        // angle *8 via three doublings: (c,s) -> (c^2 - s^2, 2cs)
        float c2 = __builtin_fmaf(c,  c,  -(s  * s )), s2 = 2.0f * c  * s;
        float c4 = __builtin_fmaf(c2, c2, -(s2 * s2)), s4 = 2.0f * c2 * s2;
        float c8 = __builtin_fmaf(c4, c4, -(s4 * s4)), s8 = 2.0f * c4 * s4;

        // (cos(phi), sin(phi)) = (x,y)/|xy|  (atan2 eliminated)
        float rxy2 = __builtin_fmaf(zx, zx, zy * zy);
        float inv  = __builtin_amdgcn_rsqf(rxy2 + 1e-30f);
        float cp = zx * inv, sp = zy * inv;
        float cp2 = __builtin_fmaf(cp,  cp,  -(sp  * sp )), sp2 = 2.0f * cp  * sp;
        float cp4 = __builtin_fmaf(cp2, cp2, -(sp2 * sp2)), sp4 = 2.0f * cp2 * sp2;
        float cp8 = __builtin_fmaf(cp4, cp4, -(sp4 * sp4)), sp8 = 2.0f * cp4 * sp4;

        // r^8 = ((r^2)^2)^2
        float r4 = r2 * r2, r8 = r4 * r4;

        float drn = __builtin_fmaf(r8 * 8.0f, dr, 1.0f);
        float nx  = __builtin_fmaf(s8 * cp8, r8, px);
        float ny  = __builtin_fmaf(s8 * sp8, r8, py);
        float nz  = __builtin_fmaf(c8,       r8, pz);

        dr = esc ? dr : drn;
        zx = esc ? zx : nx;
        zy = esc ? zy : ny;
        zz = esc ? zz : nz;
    }

    float r  = __builtin_amdgcn_sqrtf(__builtin_fmaf(zx, zx, __builtin_fmaf(zy, zy, zz * zz)));
    float ln = 0.69314718055994530942f * __builtin_amdgcn_logf(r + EPSV); // ln = ln2*log2
    float d  = 0.5f * ln * r * __builtin_amdgcn_rcpf(dr + EPSV);

    __builtin_nontemporal_store(d, dist + i);   // 302MB stream > 192MB L2 -> NT
}

// ---------------------------------------------------------------------------
// Kernel 2: attrs broadcast as a rank-1 GEMM via V_WMMA_F32_16X16X4_F32.
// Flat pattern: out[k] = attrs[k % 5]. A 256-dword tile t has
//   D_t[m][n] = attrs[(t + m + n) mod 5]   (256 = 16 = 1 mod 5)
// Expressed as D = A_lo*B_lo + A_hi*B_hi (K=8 >= 5), A one-hot, f32 exact.
// Each wave builds the 5 residue tiles once (10 WMMAs), then streams stores.
// ---------------------------------------------------------------------------
__global__ void __launch_bounds__(256)
attrs_bcast_wmma_kernel(const float* __restrict__ attrs, float* __restrict__ out,
                        long long total) {
    const int lane  = threadIdx.x & 31;
    const int m     = lane & 15;   // A: row M / B,C,D: column N
    const int khalf = lane >> 4;   // 0: K={j}, 1: K={j+2} for component j

    // B[k][n] = attrs[(k+n) % 5]; rows 0..3 -> Blo, rows 4..7 -> Bhi
    v2f Blo, Bhi;
#pragma unroll
    for (int j = 0; j < 2; ++j) {
        int klo = j + 2 * khalf;       // 0..3
        int khi = 4 + j + 2 * khalf;   // 4..7
        Blo[j] = attrs[(klo + m) % 5];
        Bhi[j] = attrs[(khi + m) % 5];
    }

    // Precompute D for all 5 tile residues
    v8f D0, D1, D2, D3, D4;
#pragma unroll
    for (int c = 0; c < 5; ++c) {
        int hot = (m + c) % 5;         // one-hot column of A row m
        v2f Alo, Ahi;
#pragma unroll
        for (int j = 0; j < 2; ++j) {
            Alo[j] = ((j + 2 * khalf)     == hot) ? 1.0f : 0.0f;
            Ahi[j] = ((4 + j + 2 * khalf) == hot) ? 1.0f : 0.0f;
        }
        v8f acc = {};
        acc = __builtin_amdgcn_wmma_f32_16x16x4_f32(false, Alo, false, Blo,
                                                    (short)0, acc, false, false);
        acc = __builtin_amdgcn_wmma_f32_16x16x4_f32(false, Ahi, false, Bhi,
                                                    (short)0, acc, false, false);
        if (c == 0) D0 = acc; else if (c == 1) D1 = acc; else if (c == 2) D2 = acc;
        else if (c == 3) D3 = acc; else D4 = acc;
    }

    // C/D layout: VGPR r, lanes 0-15 -> M=r, lanes 16-31 -> M=r+8; N = lane%16
    // flat-in-tile = 16*M + N = 16*r + 128*khalf + m
    const int loff = m + (khalf << 7);

    long long gtid   = (long long)blockIdx.x * blockDim.x + threadIdx.x;
    long long wave   = gtid >> 5;
    long long nwaves = ((long long)gridDim.x * blockDim.x) >> 5;
    long long tiles  = (total + 255) >> 8;

    for (long long t = wave; t < tiles; t += nwaves) {
        int c = (int)(t % 5);
        v8f d = (c == 0) ? D0 : (c == 1) ? D1 : (c == 2) ? D2 : (c == 3) ? D3 : D4;
        long long base = (t << 8) + loff;
        float* p = out + base;
#pragma unroll
        for (int r = 0; r < 8; ++r) {
            long long idx = base + 16 * r;
            if (idx < total)
                __builtin_nontemporal_store(d[r], p + 16 * r);
        }
    }
}

// ---------------------------------------------------------------------------
extern "C" void kernel_launch(void* const* d_in, const int* in_sizes, int n_in,
                              void* d_out, int out_size, void* d_ws, size_t ws_size,
                              hipStream_t stream) {
    const float* x     = (const float*)d_in[0];   // [N,3] f32
    const float* attrs = (const float*)d_in[1];   // [5]   f32
    float* out = (float*)d_out;                   // dist[N] ++ attrs_out[N*5]

    int n = in_sizes[0] / 3;
    float* dist      = out;
    float* attrs_out = out + n;

    int blocks = (n + 255) / 256;
    mandelbulb_dist_kernel<<<blocks, 256, 0, stream>>>(x, dist, n);

    long long total = 5ll * (long long)n;
    attrs_bcast_wmma_kernel<<<1024, 256, 0, stream>>>(attrs, attrs_out, total);
}